// backbone_msl_72739566125500
// MI455X (gfx1250) — compile-verified
//
#include <hip/hip_runtime.h>

typedef __attribute__((ext_vector_type(2))) float v2f;
typedef __attribute__((ext_vector_type(8))) float v8f;

#define S 4            // samples per workgroup
#define THREADS 256
#define NWAVES 8

// ---- LDS layout (float offsets) ----
#define OFF_X1    0        // x1  [S][25][5][18]           = 9000
#define OFF_C3    9000     // c3  [S][80][4][18] (padded M)= 23040
#define OFF_A2    32040    // a2  [S][75][4][9]            = 10800
#define OFF_RB2   42840    // rb2 [S][25][5][9]            = 4500
#define OFF_B3P   47340    // padded conv3 bias            = 80
#define OFF_BSUM  47420    // padded conv4_b+convb_b       = 128
#define LDS_FLOATS 47548   // 190,192 bytes
// Overlays:
//   front-end scratch @ OFF_C3 : xl 540 | c1 2700 | a1 1800 | rbuf 360
//   w3pad  [80][52]  = 4160  @ OFF_A2           (alive only through stage 2)
//   w4pad  [128][152]= 19456 @ 0                (stage 4 only; x1+c3 dead)
//   wcbpad [128][76] = 9728  @ 19456            (stage 4 only)

__global__ __launch_bounds__(THREADS) void backbone_fused(
    const float* __restrict__ x,
    const float* __restrict__ w1,  const float* __restrict__ b1,
    const float* __restrict__ l1w, const float* __restrict__ l1b,
    const float* __restrict__ w2,  const float* __restrict__ b2,
    const float* __restrict__ raw, const float* __restrict__ rab,
    const float* __restrict__ cvaw,const float* __restrict__ cvab,
    const float* __restrict__ w3,  const float* __restrict__ b3,
    const float* __restrict__ l2w, const float* __restrict__ l2b,
    const float* __restrict__ w4,  const float* __restrict__ b4,
    const float* __restrict__ rbw, const float* __restrict__ rbb,
    const float* __restrict__ wcb, const float* __restrict__ bcb,
    float* __restrict__ out, int Bsz)
{
    __shared__ float lds_[LDS_FLOATS];
    float* x1buf  = lds_ + OFF_X1;
    float* c3buf  = lds_ + OFF_C3;      // padded: [s][80][4][18], pitch 5760/sample
    float* a2buf  = lds_ + OFF_A2;
    float* rb2buf = lds_ + OFF_RB2;
    float* b3p    = lds_ + OFF_B3P;
    float* bsum   = lds_ + OFF_BSUM;
    // front-end scratch (overlays c3 region)
    float* xl    = c3buf;
    float* c1buf = c3buf + 540;
    float* a1buf = c3buf + 3240;
    float* rbuf  = c3buf + 5040;
    // stage-2 weights (overlay a2 region)
    float* w3p   = a2buf;               // [80][52]
    // stage-4 weights (overlay x1+c3 regions)
    float* w4p   = lds_;                // [128][152]
    float* wcbp  = lds_ + 19456;        // [128][76]

    const int tid = threadIdx.x;
    const int wg  = blockIdx.x;

    // ---- stage 0: load x; stage padded w3 / biases into LDS ----
    for (int i = tid; i < S * 135; i += THREADS) {
        long gidx = (long)wg * (S * 135) + i;
        xl[i] = (gidx < (long)Bsz * 135) ? x[gidx] : 0.f;
    }
    for (int i = tid; i < 80 * 52; i += THREADS) {
        int m = i / 52, K = i % 52;
        w3p[i] = (m < 75 && K < 50) ? w3[m * 50 + K] : 0.f;
    }
    for (int i = tid; i < 128; i += THREADS) {
        if (i < 80) b3p[i] = (i < 75) ? b3[i] : 0.f;
        bsum[i] = (i < 125) ? (b4[i] + bcb[i]) : 0.f;
    }
    __syncthreads();

    // ---- stage 1a: conv1 (pad_h=1)  c1[s,c,h,w] ----
    for (int i = tid; i < S * 675; i += THREADS) {
        int s = i / 675, r = i % 675;
        int c = r / 135, r2 = r % 135;
        int h = r2 / 27, w = r2 % 27;
        float v = b1[c];
#pragma unroll
        for (int kh = 0; kh < 3; ++kh) {
            int hh = h + kh - 1;
            if (hh >= 0 && hh < 5) v += xl[s * 135 + hh * 27 + w] * w1[c * 3 + kh];
        }
        c1buf[i] = v;
    }
    __syncthreads();

    // ---- stage 1b: l1 + relu ; ra ----
    for (int i = tid; i < S * 450; i += THREADS) {
        int s = i / 450, r = i % 450;
        int c = r / 90, r2 = r % 90;
        int h = r2 / 18, j = r2 % 18;
        float v = l1b[j];
#pragma unroll
        for (int w = 0; w < 27; ++w)
            v += c1buf[s * 675 + c * 135 + h * 27 + w] * l1w[j * 27 + w];
        a1buf[i] = v > 0.f ? v : 0.f;
    }
    for (int i = tid; i < S * 90; i += THREADS) {
        int s = i / 90, r = i % 90;
        int h = r / 18, j = r % 18;
        float v = rab[j];
#pragma unroll
        for (int w = 0; w < 27; ++w)
            v += xl[s * 135 + h * 27 + w] * raw[j * 27 + w];
        rbuf[i] = v;
    }
    __syncthreads();

    // ---- stage 1c: conv2 + conva -> x1[s,o,h,j] ----
    for (int i = tid; i < S * 2250; i += THREADS) {
        int s = i / 2250, r = i % 2250;
        int o = r / 90, r2 = r % 90;
        int h = r2 / 18, j = r2 % 18;
        float v = b2[o];
#pragma unroll
        for (int ic = 0; ic < 5; ++ic)
#pragma unroll
            for (int kh = 0; kh < 3; ++kh) {
                int hh = h + kh - 1;
                if (hh >= 0 && hh < 5)
                    v += a1buf[s * 450 + ic * 90 + hh * 18 + j] * w2[o * 15 + ic * 3 + kh];
            }
        v += cvaw[o] * rbuf[s * 90 + h * 18 + j] + cvab[o];
        x1buf[i] = v;
    }
    __syncthreads();

    const int lane  = tid & 31;
    const int wave  = tid >> 5;
    const int ln15  = lane & 15;
    const int khalf = lane >> 4;

    // ---- stage 2: conv3 WMMA GEMM.  M=80(pad), N=288, K=50 (13 steps) ----
    for (int t = wave; t < 5 * 18; t += NWAVES) {
        int mt = t / 18, nt = t % 18;
        int abase = (mt * 16 + ln15) * 52;
        int col = nt * 16 + ln15;
        int s = col / 72, rem = col % 72;
        int h = rem / 18, w = rem % 18;
        int bbase = s * 2250 + h * 18 + w + khalf * 90;  // x1 @ (i=khalf, kh=0)
        v8f acc;
#pragma unroll
        for (int r = 0; r < 8; ++r)
            acc[r] = b3p[mt * 16 + r + khalf * 8];
#pragma unroll
        for (int kb = 0; kb < 13; ++kb) {
            int K0 = kb * 4 + khalf * 2;
            v2f va, vb;
            va.x = w3p[abase + K0];
            va.y = w3p[abase + K0 + 1];
            vb.x = x1buf[bbase + kb * 180];        // (i=2kb+khalf, kh=0)
            vb.y = x1buf[bbase + kb * 180 + 18];   // kh=1
            acc = __builtin_amdgcn_wmma_f32_16x16x4_f32(
                false, va, false, vb, (short)0, acc, false, false);
        }
#pragma unroll
        for (int r = 0; r < 8; ++r) {
            int mo = mt * 16 + r + khalf * 8;      // < 80, padded store
            c3buf[s * 5760 + mo * 72 + h * 18 + w] = acc[r];
        }
    }
    __syncthreads();

    // ---- stage 3: l2+relu -> a2 ; rb -> rb2 ----
    for (int i = tid; i < S * 2700; i += THREADS) {
        int s = i / 2700, r = i % 2700;
        int o = r / 36, r2 = r % 36;
        int h = r2 / 9, j = r2 % 9;
        float v = l2b[j];
#pragma unroll
        for (int w = 0; w < 18; ++w)
            v += c3buf[s * 5760 + o * 72 + h * 18 + w] * l2w[j * 18 + w];
        a2buf[i] = v > 0.f ? v : 0.f;
    }
    for (int i = tid; i < S * 1125; i += THREADS) {
        int s = i / 1125, r = i % 1125;
        int ic = r / 45, r2 = r % 45;
        int h = r2 / 9, j = r2 % 9;
        float v = rbb[j];
#pragma unroll
        for (int w = 0; w < 18; ++w)
            v += x1buf[s * 2250 + ic * 90 + h * 18 + w] * rbw[j * 18 + w];
        rb2buf[i] = v;
    }
    __syncthreads();

    // ---- stage 3.5: stage padded w4 / wcb into (now dead) x1+c3 regions ----
    for (int i = tid; i < 128 * 152; i += THREADS) {
        int m = i / 152, K = i % 152;
        w4p[i] = (m < 125 && K < 150) ? w4[m * 150 + K] : 0.f;
    }
    for (int i = tid; i < 128 * 76; i += THREADS) {
        int m = i / 76, K = i % 76;
        wcbp[i] = (m < 125 && K < 75) ? wcb[m * 75 + K] : 0.f;
    }
    __syncthreads();

    // ---- stage 4: conv4 (38 steps) + convb (19 steps), fused accumulator ----
    for (int t = wave; t < 8 * 7; t += NWAVES) {
        int mt = t / 7, nt = t % 7;
        int row = mt * 16 + ln15;
        int abase4 = row * 152;
        int abaseb = row * 76;
        int col = nt * 16 + ln15;
        bool cval = col < S * 27;
        int cc = cval ? col : 0;                 // clamp: safe LDS addr for pad lanes
        int s = cc / 27, rem = cc % 27;
        int h = rem / 9, w = rem % 9;
        int b4base = s * 2700 + h * 9 + w + khalf * 36;  // a2  @ (i=khalf, kh=0)
        int bbbase = s * 1125 + h * 9 + w;               // rb2 @ (i=0,  kh=0)
        v8f acc;
#pragma unroll
        for (int r = 0; r < 8; ++r)
            acc[r] = bsum[mt * 16 + r + khalf * 8];
#pragma unroll
        for (int kb = 0; kb < 38; ++kb) {
            int K0 = kb * 4 + khalf * 2;
            v2f va, vb;
            va.x = w4p[abase4 + K0];
            va.y = w4p[abase4 + K0 + 1];
            vb.x = a2buf[b4base + kb * 72];       // (i=2kb+khalf, kh=0)
            vb.y = a2buf[b4base + kb * 72 + 9];   // kh=1
            acc = __builtin_amdgcn_wmma_f32_16x16x4_f32(
                false, va, false, vb, (short)0, acc, false, false);
        }
#pragma unroll
        for (int kb = 0; kb < 19; ++kb) {
            int K0 = kb * 4 + khalf * 2;
            int K1 = K0 + 1;
            int i0 = K0 / 3, kh0 = K0 % 3;        // compile-time after unroll
            int i1 = K1 / 3, kh1 = K1 % 3;
            v2f va, vb;
            va.x = wcbp[abaseb + K0];
            va.y = wcbp[abaseb + K1];
            vb.x = rb2buf[bbbase + i0 * 45 + kh0 * 9];
            vb.y = rb2buf[bbbase + i1 * 45 + kh1 * 9];
            acc = __builtin_amdgcn_wmma_f32_16x16x4_f32(
                false, va, false, vb, (short)0, acc, false, false);
        }
        if (cval) {
            long gs = (long)wg * S + s;
            if (gs < Bsz) {
#pragma unroll
                for (int r = 0; r < 8; ++r) {
                    int mo = mt * 16 + r + khalf * 8;
                    if (mo < 125) out[gs * 3375 + mo * 27 + h * 9 + w] = acc[r];
                }
            }
        }
    }
}

extern "C" void kernel_launch(void* const* d_in, const int* in_sizes, int n_in,
                              void* d_out, int out_size, void* d_ws, size_t ws_size,
                              hipStream_t stream) {
    (void)n_in; (void)out_size; (void)d_ws; (void)ws_size;
    const float* x    = (const float*)d_in[0];
    const float* w1   = (const float*)d_in[1];
    const float* b1   = (const float*)d_in[2];
    const float* l1w  = (const float*)d_in[3];
    const float* l1b  = (const float*)d_in[4];
    const float* w2   = (const float*)d_in[5];
    const float* b2   = (const float*)d_in[6];
    const float* raw  = (const float*)d_in[7];
    const float* rab  = (const float*)d_in[8];
    const float* cvaw = (const float*)d_in[9];
    const float* cvab = (const float*)d_in[10];
    const float* w3   = (const float*)d_in[11];
    const float* b3   = (const float*)d_in[12];
    const float* l2w  = (const float*)d_in[13];
    const float* l2b  = (const float*)d_in[14];
    const float* w4   = (const float*)d_in[15];
    const float* b4   = (const float*)d_in[16];
    const float* rbw  = (const float*)d_in[17];
    const float* rbb  = (const float*)d_in[18];
    const float* wcb  = (const float*)d_in[19];
    const float* bcb  = (const float*)d_in[20];

    int Bsz  = in_sizes[0] / 135;
    int nblk = (Bsz + S - 1) / S;
    backbone_fused<<<nblk, THREADS, 0, stream>>>(
        x, w1, b1, l1w, l1b, w2, b2, raw, rab, cvaw, cvab,
        w3, b3, l2w, l2b, w4, b4, rbw, rbb, wcb, bcb,
        (float*)d_out, Bsz);
}